// DeformConv2dPack_84404697301198
// MI455X (gfx1250) — compile-verified
//
#include <hip/hip_runtime.h>
#include <hip/hip_bf16.h>

typedef float v2f __attribute__((ext_vector_type(2)));
typedef float v8f __attribute__((ext_vector_type(8)));

#define BATCH 4
#define CIN   64
#define COUT  64
#define HH    128
#define WW    128
#define HW    (HH * WW)          // 16384
#define K2    9
#define KDIM  (CIN * K2)         // 576
#define OCOFF 18                 // 2*K2 offset channels

// ---------------------------------------------------------------------------
// Kernel 1: offset = conv3x3(x, w_offset) + b_offset   (64 -> 18 channels)
// One thread per output pixel; weights staged in LDS as [c][tap][oc] so the
// inner oc loop is an LDS broadcast (all lanes read the same address).
// ---------------------------------------------------------------------------
__global__ __launch_bounds__(256) void offset_conv_kernel(
    const float* __restrict__ x, const float* __restrict__ w_off,
    const float* __restrict__ b_off, float* __restrict__ off_out)
{
    __shared__ float sw[CIN * K2 * OCOFF];   // [c][t][oc] = 10368 floats
    __shared__ float sb[OCOFF];
    const int tid = threadIdx.x;

    for (int i = tid; i < CIN * K2 * OCOFF; i += 256) {
        int oc = i % OCOFF;
        int t  = (i / OCOFF) % K2;
        int c  = i / (OCOFF * K2);
        sw[i] = w_off[oc * KDIM + c * K2 + t];   // w_offset is (18,64,3,3)
    }
    if (tid < OCOFF) sb[tid] = b_off[tid];
    __syncthreads();

    const int pix = blockIdx.x * 256 + tid;       // 0 .. B*H*W-1
    const int b   = pix >> 14;
    const int rem = pix & (HW - 1);
    const int h   = rem >> 7;
    const int w   = rem & (WW - 1);

    float acc[OCOFF];
#pragma unroll
    for (int oc = 0; oc < OCOFF; ++oc) acc[oc] = sb[oc];

    const float* xb = x + b * (CIN * HW);
    for (int c = 0; c < CIN; ++c) {
        float xv[K2];
#pragma unroll
        for (int t = 0; t < K2; ++t) {
            int hy = h + t / 3 - 1;
            int wx = w + t % 3 - 1;
            bool ok = (hy >= 0) & (hy < HH) & (wx >= 0) & (wx < WW);
            xv[t] = ok ? xb[c * HW + hy * WW + wx] : 0.0f;
        }
        const float* wc = &sw[c * (K2 * OCOFF)];
#pragma unroll
        for (int t = 0; t < K2; ++t) {
#pragma unroll
            for (int oc = 0; oc < OCOFF; ++oc)
                acc[oc] = fmaf(xv[t], wc[t * OCOFF + oc], acc[oc]);
        }
    }

    float* ob = off_out + b * (OCOFF * HW) + h * WW + w;
#pragma unroll
    for (int oc = 0; oc < OCOFF; ++oc) ob[oc * HW] = acc[oc];
}

// ---------------------------------------------------------------------------
// Kernel 2: deformable conv as implicit GEMM on V_WMMA_F32_16X16X4_F32.
// Workgroup = 128 threads (4 waves). M-tile = 16 contiguous pixels in one
// image row; wave w owns output channels [16w, 16w+16).
// Per tap: sample 16x64 A-tile into LDS, then 16 WMMA k-steps (4 ch each).
// K-flattening matches the reference: k = c*9 + tap on both A and B.
// ---------------------------------------------------------------------------
__global__ __launch_bounds__(128) void deform_conv_wmma_kernel(
    const float* __restrict__ x, const float* __restrict__ off,
    const float* __restrict__ wd, const float* __restrict__ bd,
    float* __restrict__ out)
{
    __shared__ float smp[CIN * 16];     // sampled A-tile, [c][m]
    __shared__ float sout[16 * COUT];   // D-tile staging, [m][n]

    const int tid  = threadIdx.x;
    const int lane = tid & 31;
    const int wave = tid >> 5;

    const int pix0 = blockIdx.x * 16;
    const int b    = pix0 >> 14;
    const int rem  = pix0 & (HW - 1);
    const int h    = rem >> 7;
    const int w0   = rem & (WW - 1);    // tile never crosses a row (128 % 16 == 0)

    // sampling mapping: thread -> (pixel m, 8 channels)
    const int sm = tid & 15;            // pixel in tile (lane-contiguous -> coalesced)
    const int c0 = (tid >> 4) * 8;      // channel group base

    // WMMA fragment mapping (32-bit A: vgpr0 = K 0/2 by lane-half, vgpr1 = K 1/3)
    const int hi = lane >> 4;           // lane-half
    const int mm = lane & 15;           // A row / B,D column within wave tile
    const int nn = (wave << 4) + mm;    // global output channel for B/D column

    const float* xb   = x + b * (CIN * HW);
    const float* offb = off + b * (OCOFF * HW) + h * WW + (w0 + sm);
    const float* wdn  = wd + nn * KDIM;  // w_deform row for this lane's cout

    v8f acc = {};

    for (int q = 0; q < K2; ++q) {
        // ---- bilinear sampling parameters for (pixel sm, tap q) ----
        float dy = offb[(2 * q) * HW];
        float dx = offb[(2 * q + 1) * HW];
        float py = (float)(h - 1 + q / 3) + dy;
        float px = (float)(w0 + sm - 1 + q % 3) + dx;
        float y0f = floorf(py), x0f = floorf(px);
        int   y0 = (int)y0f,   x0 = (int)x0f;
        float wy1 = py - y0f, wy0 = 1.0f - wy1;
        float wx1 = px - x0f, wx0 = 1.0f - wx1;

        int   cidx[4];
        float cw[4];
#pragma unroll
        for (int cor = 0; cor < 4; ++cor) {
            int iy = cor >> 1, ix = cor & 1;
            int yy = y0 + iy, xx = x0 + ix;
            bool ok = (yy >= 0) & (yy < HH) & (xx >= 0) & (xx < WW);
            int yc = min(max(yy, 0), HH - 1);
            int xc = min(max(xx, 0), WW - 1);
            cidx[cor] = yc * WW + xc;
            cw[cor] = (iy ? wy1 : wy0) * (ix ? wx1 : wx0) * (ok ? 1.0f : 0.0f);
        }

        if (q) __syncthreads();   // previous WMMA phase done reading smp
#pragma unroll
        for (int cc = 0; cc < 8; ++cc) {
            int c = c0 + cc;
            const float* xc = xb + c * HW;
            float v = cw[0] * xc[cidx[0]] + cw[1] * xc[cidx[1]]
                    + cw[2] * xc[cidx[2]] + cw[3] * xc[cidx[3]];
            smp[c * 16 + sm] = v;
        }
        __syncthreads();

        // ---- 16 WMMA k-steps over the 64 channels of this tap ----
        const float* wq = wdn + q;      // + c*9 per channel
#pragma unroll
        for (int kk = 0; kk < 16; ++kk) {
            int kb = kk * 4 + 2 * hi;   // channel index for the .x element
            v2f a, bb;
            a.x  = smp[kb * 16 + mm];
            a.y  = smp[(kb + 1) * 16 + mm];
            bb.x = wq[kb * K2];
            bb.y = wq[(kb + 1) * K2];
            acc = __builtin_amdgcn_wmma_f32_16x16x4_f32(
                false, a, false, bb, (short)0, acc, false, false);
        }
    }

    // ---- transpose D through LDS, then coalesced row stores (+bias) ----
    __syncthreads();
#pragma unroll
    for (int r = 0; r < 8; ++r) {
        int M = r + 8 * hi;             // D: lanes 0-15 -> M=vgpr, 16-31 -> M=vgpr+8
        sout[M * COUT + nn] = acc[r];
    }
    __syncthreads();

    const int oc = tid >> 1;
    const int mh = (tid & 1) * 8;
    float bias = bd[oc];
    float* ob = out + b * (COUT * HW) + oc * HW + h * WW + w0 + mh;
#pragma unroll
    for (int i = 0; i < 8; ++i) ob[i] = sout[(mh + i) * COUT + oc] + bias;
}

// ---------------------------------------------------------------------------
extern "C" void kernel_launch(void* const* d_in, const int* in_sizes, int n_in,
                              void* d_out, int out_size, void* d_ws, size_t ws_size,
                              hipStream_t stream) {
    const float* x     = (const float*)d_in[0];
    const float* w_off = (const float*)d_in[1];
    const float* b_off = (const float*)d_in[2];
    const float* w_def = (const float*)d_in[3];
    const float* b_def = (const float*)d_in[4];
    float* out = (float*)d_out;
    float* off = (float*)d_ws;   // BATCH*18*H*W floats = 4.7 MB scratch

    offset_conv_kernel<<<(BATCH * HW) / 256, 256, 0, stream>>>(x, w_off, b_off, off);
    deform_conv_wmma_kernel<<<(BATCH * HW) / 16, 128, 0, stream>>>(x, off, w_def, b_def, out);
}